// SimVQ_62783831933260
// MI455X (gfx1250) — compile-verified
//
#include <hip/hip_runtime.h>
#include <stdint.h>

typedef __attribute__((ext_vector_type(2))) float v2f;
typedef __attribute__((ext_vector_type(8))) float v8f;

#define N_E     8192
#define E_DIM   32
#define M_TOT   16384      // 2*8*32*32 latent vectors
#define HALF_Z  262144     // 32*8192 elements per batch slice of z
#define CHUNK_N 128        // codebook rows staged in LDS per iteration
#define LDS_STRIDE 33      // 32 + 1 pad -> conflict-free column access

// Map float to uint so that uint compare == float compare (total order).
static __device__ __forceinline__ uint32_t f32_orderable(float f) {
    uint32_t u = __float_as_uint(f);
    return u ^ ((uint32_t)((int32_t)u >> 31) | 0x80000000u);
}

static __device__ __forceinline__ uint64_t shfl_xor_u64(uint64_t v, int mask) {
    int lo = (int)(uint32_t)(v & 0xffffffffull);
    int hi = (int)(uint32_t)(v >> 32);
    lo = __shfl_xor(lo, mask, 32);
    hi = __shfl_xor(hi, mask, 32);
    return ((uint64_t)(uint32_t)hi << 32) | (uint32_t)(uint32_t)lo;
}

// ---------------------------------------------------------------------------
// Kernel 1: cb[n][k] = dot(emb[n], proj_w[k]) + proj_b[k];  cbn2[n] = |cb[n]|^2
// One wave (32 lanes) per codebook row: lane = output column k.
// ---------------------------------------------------------------------------
__global__ void __launch_bounds__(256)
simvq_cb_kernel(const float* __restrict__ emb, const float* __restrict__ pw,
                const float* __restrict__ pb, float* __restrict__ cb,
                float* __restrict__ cbn2) {
    int tid  = threadIdx.x;
    int wave = tid >> 5;
    int lane = tid & 31;
    int n = blockIdx.x * 8 + wave;
    int k = lane;
    const float* er = emb + (size_t)n * E_DIM;
    const float* wr = pw  + (size_t)k * E_DIM;
    float acc = pb[k];
#pragma unroll
    for (int j = 0; j < E_DIM; ++j) acc += er[j] * wr[j];
    cb[(size_t)n * E_DIM + k] = acc;
    float sq = acc * acc;
#pragma unroll
    for (int off = 16; off >= 1; off >>= 1) sq += __shfl_xor(sq, off, 32);
    if (lane == 0) cbn2[n] = sq;
}

// ---------------------------------------------------------------------------
// Kernel 2: argmin_n (|cb_n|^2 - 2 * z_m . cb_n)  via V_WMMA_F32_16X16X4_F32.
// Block = 8 waves; each wave owns a 16-row A tile (z rows), K=32 chained as
// 8 WMMAs of K=4. Codebook staged in LDS in 128-row chunks shared by 8 waves.
//
// Argmin strategy: each lane keeps a private running (dist, n) min for its
// column class (branchless cndmask select; within a lane n is increasing so
// strict '<' preserves first-index ties). One packed-key u64 butterfly
// min-reduce across the 16 N-lanes AFTER the loop gives min-dist-then-min-n,
// matching jnp.argmin exactly — zero cross-lane ops in the hot loop.
// ---------------------------------------------------------------------------
__global__ void __launch_bounds__(256)
simvq_argmin_kernel(const float* __restrict__ z, const float* __restrict__ cb,
                    const float* __restrict__ cbn2, int* __restrict__ out_idx) {
    __shared__ float lds_cb[CHUNK_N * LDS_STRIDE];
    __shared__ float lds_n2[CHUNK_N];

    int tid  = threadIdx.x;
    int wave = tid >> 5;
    int lane = tid & 31;
    int lo   = lane & 15;     // M row within tile (A), N column (B/C)
    int hi   = lane >> 4;     // K half select (A/B), row +8 select (C)
    int k0   = hi * 2;

    int mbase = (blockIdx.x * 8 + wave) * 16;
    int m  = mbase + lo;
    int bb = m >> 13;         // batch index
    int r  = m & 8191;        // t*1024 + h*32 + w
    const float* zrow = z + (size_t)bb * HALF_Z + r;   // stride 8192 over k

    // Load the wave's A fragments once (reused for all 8192 codes).
    v2f a[8];
#pragma unroll
    for (int c = 0; c < 8; ++c) {
        int k = 4 * c + k0;
        a[c].x = zrow[(size_t)k * 8192];
        a[c].y = zrow[(size_t)(k + 1) * 8192];
    }

    float best[8];
    int   bn[8];
#pragma unroll
    for (int j = 0; j < 8; ++j) { best[j] = 3.4e38f; bn[j] = 0; }

    for (int nbase = 0; nbase < N_E; nbase += CHUNK_N) {
        __syncthreads();
        // Cooperative stage of 128x32 codebook chunk into padded LDS.
#pragma unroll
        for (int j = 0; j < 4; ++j) {
            int i4  = tid + j * 256;        // float4 slot, 0..1023
            int row = i4 >> 3;              // 8 float4 per 32-col row
            int col = (i4 & 7) * 4;
            float4 v = *reinterpret_cast<const float4*>(
                cb + (size_t)(nbase + row) * E_DIM + col);
            float* dst = &lds_cb[row * LDS_STRIDE + col];
            dst[0] = v.x; dst[1] = v.y; dst[2] = v.z; dst[3] = v.w;
        }
        if (tid < CHUNK_N) lds_n2[tid] = cbn2[nbase + tid];
        if (nbase + CHUNK_N < N_E)      // warm L2/L0 for the next chunk
            __builtin_prefetch(cb + (size_t)(nbase + CHUNK_N) * E_DIM + tid * 16, 0, 3);
        __syncthreads();

#pragma unroll
        for (int s = 0; s < CHUNK_N / 16; ++s) {
            const float* brow = &lds_cb[(s * 16 + lo) * LDS_STRIDE];
            v8f acc = {};
#pragma unroll
            for (int c = 0; c < 8; ++c) {
                int k = 4 * c + k0;
                v2f bf;
                bf.x = brow[k];
                bf.y = brow[k + 1];
                acc = __builtin_amdgcn_wmma_f32_16x16x4_f32(
                    /*neg_a=*/false, a[c], /*neg_b=*/false, bf,
                    /*c_mod=*/(short)0, acc, /*reuse_a=*/false, /*reuse_b=*/false);
            }
            int   nidx = nbase + s * 16 + lo;
            float n2   = lds_n2[s * 16 + lo];
#pragma unroll
            for (int j = 0; j < 8; ++j) {
                float d = __builtin_fmaf(-2.0f, acc[j], n2);  // |z|^2 is row-const
                bool lt = d < best[j];                        // branchless select
                best[j] = lt ? d : best[j];
                bn[j]   = lt ? nidx : bn[j];
            }
        }
    }

    // One-time cross-lane reduction over the 16 N-lanes per row:
    // packed key = (orderable(dist) << 32) | n  -> u64 min == (min d, min n).
#pragma unroll
    for (int j = 0; j < 8; ++j) {
        uint64_t key = ((uint64_t)f32_orderable(best[j]) << 32) | (uint32_t)bn[j];
#pragma unroll
        for (int off = 1; off < 16; off <<= 1) {
            uint64_t ok = shfl_xor_u64(key, off);
            key = (ok < key) ? ok : key;
        }
        bn[j] = (int)(uint32_t)(key & 0xffffffffull);
    }

    if (lo == 0) {            // lanes 0 and 16 carry converged results
#pragma unroll
        for (int j = 0; j < 8; ++j)
            out_idx[mbase + j + 8 * hi] = bn[j];
    }
}

// ---------------------------------------------------------------------------
// Kernel 3: z_q gather back to (b,c,t,h,w), index emit, per-block loss partial.
// ---------------------------------------------------------------------------
__global__ void __launch_bounds__(256)
simvq_gather_kernel(const float* __restrict__ z, const float* __restrict__ cb,
                    const int* __restrict__ idx, float* __restrict__ out,
                    float* __restrict__ partial) {
    __shared__ float red[256];
    int m  = blockIdx.x * 256 + threadIdx.x;
    int bb = m >> 13;
    int r  = m & 8191;
    int id = idx[m];
    const float* crow = cb + (size_t)id * E_DIM;
    const float* zrow = z   + (size_t)bb * HALF_Z + r;
    float*       qrow = out + (size_t)bb * HALF_Z + r;
    float s = 0.0f;
#pragma unroll
    for (int k = 0; k < E_DIM; ++k) {
        float cv  = crow[k];
        float dlt = cv - zrow[(size_t)k * 8192];
        s += dlt * dlt;
        qrow[(size_t)k * 8192] = cv;      // STE forward value == z_q
    }
    out[131072 + 1 + m] = (float)id;      // indices region (float dtype d_out)
    red[threadIdx.x] = s;
    __syncthreads();
    for (int off = 128; off >= 1; off >>= 1) {
        if (threadIdx.x < off) red[threadIdx.x] += red[threadIdx.x + off];
        __syncthreads();
    }
    if (threadIdx.x == 0) partial[blockIdx.x] = red[0];
}

// ---------------------------------------------------------------------------
// Kernel 4: loss = 1.25 * sum(partials) / (M*E_DIM)   (deterministic)
// ---------------------------------------------------------------------------
__global__ void simvq_loss_kernel(const float* __restrict__ partial,
                                  float* __restrict__ out) {
    __shared__ float red[64];
    int t = threadIdx.x;
    red[t] = partial[t];
    __syncthreads();
    for (int off = 32; off >= 1; off >>= 1) {
        if (t < off) red[t] += red[t + off];
        __syncthreads();
    }
    if (t == 0) out[131072] = red[0] * (1.25f / (float)(M_TOT * E_DIM));
}

// ---------------------------------------------------------------------------
extern "C" void kernel_launch(void* const* d_in, const int* in_sizes, int n_in,
                              void* d_out, int out_size, void* d_ws, size_t ws_size,
                              hipStream_t stream) {
    const float* z   = (const float*)d_in[0];
    const float* emb = (const float*)d_in[1];
    const float* pw  = (const float*)d_in[2];
    const float* pb  = (const float*)d_in[3];
    float* out = (float*)d_out;
    float* ws  = (float*)d_ws;

    float* cb      = ws;                           // 262144 f32 (1 MB)
    float* cbn2    = ws + 262144;                  //   8192 f32
    int*   idx     = (int*)(ws + 262144 + 8192);   //  16384 i32
    float* partial = ws + 262144 + 8192 + 16384;   //     64 f32

    simvq_cb_kernel    <<<N_E / 8,        256, 0, stream>>>(emb, pw, pb, cb, cbn2);
    simvq_argmin_kernel<<<M_TOT / (8*16), 256, 0, stream>>>(z, cb, cbn2, idx);
    simvq_gather_kernel<<<M_TOT / 256,    256, 0, stream>>>(z, cb, idx, out, partial);
    simvq_loss_kernel  <<<1,              64,  0, stream>>>(partial, out);
}